// GCN_30502857736244
// MI455X (gfx1250) — compile-verified
//
#include <hip/hip_runtime.h>
#include <hip/hip_bf16.h>

typedef __attribute__((ext_vector_type(2))) float v2f;
typedef __attribute__((ext_vector_type(4))) float v4f;
typedef __attribute__((ext_vector_type(8))) float v8f;
typedef __attribute__((ext_vector_type(2))) int   v2i;

#define N_NODES 100000
#define N_EDGES 3200000
#define C_DIM 512
#define H_DIM 256
#define F_DIM 64

// ---------------------------------------------------------------------------
// Pack B[K x Ncol] into per-tile WMMA fragment order:
//   Bp[(tile*(K/4) + kk)*32 + lane] = { B[k+2h][n0+l15], B[k+2h+1][n0+l15] }
// so the GEMM fetches one coalesced b64 per fragment.
// ---------------------------------------------------------------------------
__global__ void packB_kernel(const float* __restrict__ B, v2f* __restrict__ Bp,
                             int K, int Ncol) {
  const int total = (Ncol / 16) * (K / 4) * 32;
  int idx = blockIdx.x * blockDim.x + threadIdx.x;
  if (idx >= total) return;
  const int K4   = K >> 2;
  const int lane = idx & 31;
  const int kk   = (idx >> 5) % K4;
  const int t    = idx / (K4 * 32);
  const int h    = lane >> 4;
  const int l15  = lane & 15;
  const int k    = kk * 4 + 2 * h;
  v2f b;
  b.x = B[(size_t)k * Ncol + t * 16 + l15];
  b.y = B[(size_t)(k + 1) * Ncol + t * 16 + l15];
  Bp[idx] = b;
}

// ---------------------------------------------------------------------------
// Register-blocked GEMM: each wave computes a 32x64 block of C using fp32
// WMMA 16x16x4 (2 M-tiles x 4 N-tiles => 8 WMMAs per K-step of 4).
// grid = (M/32, Ncol/64), block = 1 wave.
// ---------------------------------------------------------------------------
__global__ __launch_bounds__(32) void gemm_wmma_v2(
    const float* __restrict__ A, const v2f* __restrict__ Bp,
    float* __restrict__ C, int K, int Ncol) {
  const int lane = threadIdx.x;
  const int l15  = lane & 15;
  const int half = lane >> 4;               // A/B fragment K-half per ISA layout
  const int K4   = K >> 2;

  const size_t m0    = (size_t)blockIdx.x * 32;
  const float* arow0 = A + (m0 + l15) * (size_t)K + 2 * half;
  const float* arow1 = arow0 + (size_t)16 * K;
  const v2f*   bp    = Bp + (size_t)(blockIdx.y * 4) * K4 * 32 + lane;

  v8f acc[2][4];
#pragma unroll
  for (int i = 0; i < 2; ++i)
#pragma unroll
    for (int j = 0; j < 4; ++j) acc[i][j] = (v8f){0.f,0.f,0.f,0.f,0.f,0.f,0.f,0.f};

#pragma unroll 2
  for (int kk = 0; kk < K4; ++kk) {
    const int k = kk * 4;
    v2f a0; a0.x = arow0[k]; a0.y = arow0[k + 1];
    v2f a1; a1.x = arow1[k]; a1.y = arow1[k + 1];
    v2f b0 = bp[(size_t)(0 * K4 + kk) * 32];
    v2f b1 = bp[(size_t)(1 * K4 + kk) * 32];
    v2f b2 = bp[(size_t)(2 * K4 + kk) * 32];
    v2f b3 = bp[(size_t)(3 * K4 + kk) * 32];
    acc[0][0] = __builtin_amdgcn_wmma_f32_16x16x4_f32(false, a0, false, b0, (short)0, acc[0][0], false, false);
    acc[0][1] = __builtin_amdgcn_wmma_f32_16x16x4_f32(false, a0, false, b1, (short)0, acc[0][1], false, false);
    acc[0][2] = __builtin_amdgcn_wmma_f32_16x16x4_f32(false, a0, false, b2, (short)0, acc[0][2], false, false);
    acc[0][3] = __builtin_amdgcn_wmma_f32_16x16x4_f32(false, a0, false, b3, (short)0, acc[0][3], false, false);
    acc[1][0] = __builtin_amdgcn_wmma_f32_16x16x4_f32(false, a1, false, b0, (short)0, acc[1][0], false, false);
    acc[1][1] = __builtin_amdgcn_wmma_f32_16x16x4_f32(false, a1, false, b1, (short)0, acc[1][1], false, false);
    acc[1][2] = __builtin_amdgcn_wmma_f32_16x16x4_f32(false, a1, false, b2, (short)0, acc[1][2], false, false);
    acc[1][3] = __builtin_amdgcn_wmma_f32_16x16x4_f32(false, a1, false, b3, (short)0, acc[1][3], false, false);
  }

  // C/D layout: VGPR r -> lanes 0-15: M=r, N=lane; lanes 16-31: M=8+r, N=lane-16
  const int n0 = blockIdx.y * 64;
#pragma unroll
  for (int i = 0; i < 2; ++i) {
    float* crow = C + (m0 + i * 16 + 8 * half) * (size_t)Ncol + n0 + l15;
#pragma unroll
    for (int j = 0; j < 4; ++j)
#pragma unroll
      for (int r = 0; r < 8; ++r) crow[(size_t)r * Ncol + j * 16] = acc[i][j][r];
  }
}

// ---------------------------------------------------------------------------
// CSR build: histogram -> exclusive scan -> scatter into packed (col,val)
// ---------------------------------------------------------------------------
__global__ void hist_kernel(const int* __restrict__ row, int* __restrict__ counts, int E) {
  int i = blockIdx.x * blockDim.x + threadIdx.x;
  if (i < E) atomicAdd(&counts[row[i]], 1);
}

__global__ __launch_bounds__(1024) void exclusive_scan_kernel(
    const int* __restrict__ in, int* __restrict__ out, int n) {
  __shared__ int sdata[1024];
  __shared__ int carry;
  const int tid = threadIdx.x;
  if (tid == 0) carry = 0;
  __syncthreads();
  for (int base = 0; base < n; base += 1024) {
    int i = base + tid;
    int v = (i < n) ? in[i] : 0;
    sdata[tid] = v;
    __syncthreads();
    for (int off = 1; off < 1024; off <<= 1) {
      int t = (tid >= off) ? sdata[tid - off] : 0;
      __syncthreads();
      sdata[tid] += t;
      __syncthreads();
    }
    int incl = sdata[tid];
    if (i < n) out[i] = incl - v + carry;   // exclusive prefix
    __syncthreads();
    if (tid == 1023) carry += sdata[1023];
    __syncthreads();
  }
  if (tid == 0) out[n] = carry;             // total = E
}

__global__ void scatter_kernel(const int* __restrict__ row, const int* __restrict__ col,
                               const float* __restrict__ val,
                               const int* __restrict__ rowptr, int* __restrict__ fill,
                               v2i* __restrict__ epair, int E) {
  int i = blockIdx.x * blockDim.x + threadIdx.x;
  if (i < E) {
    int r = row[i];
    int pos = rowptr[r] + atomicAdd(&fill[r], 1);
    v2i ed;
    ed.x = col[i];
    ed.y = __float_as_int(val[i]);
    epair[pos] = ed;
  }
}

// ---------------------------------------------------------------------------
// SpMM (CSR, D=256) + fused relu: one wave per row. Lanes hold 4 consecutive
// columns (b128 gathers); feature matrix is L2-resident.
// ---------------------------------------------------------------------------
__global__ __launch_bounds__(256) void spmm_relu256(
    const int* __restrict__ rowptr, const v2i* __restrict__ epair,
    const float* __restrict__ Zin, float* __restrict__ Zout, int nrows) {
  const int lane = threadIdx.x & 31;
  const int wave = threadIdx.x >> 5;
  const int r = blockIdx.x * 8 + wave;
  if (r >= nrows) return;

  v4f acc0 = (v4f){0.f, 0.f, 0.f, 0.f};
  v4f acc1 = (v4f){0.f, 0.f, 0.f, 0.f};

  const int e0 = rowptr[r], e1 = rowptr[r + 1];
  for (int e = e0; e < e1; ++e) {
    const v2i   ed = epair[e];
    const float v  = __int_as_float(ed.y);
    const v4f*  z  = (const v4f*)(Zin + (size_t)ed.x * 256);
    acc0 += v * z[lane];
    acc1 += v * z[lane + 32];
  }

  v4f r0, r1;
  r0.x = fmaxf(acc0.x, 0.f); r0.y = fmaxf(acc0.y, 0.f);
  r0.z = fmaxf(acc0.z, 0.f); r0.w = fmaxf(acc0.w, 0.f);
  r1.x = fmaxf(acc1.x, 0.f); r1.y = fmaxf(acc1.y, 0.f);
  r1.z = fmaxf(acc1.z, 0.f); r1.w = fmaxf(acc1.w, 0.f);

  v4f* o = (v4f*)(Zout + (size_t)r * 256);
  o[lane]      = r0;
  o[lane + 32] = r1;
}

// SpMM (D=64) + fused row softmax: lanes hold 2 consecutive columns.
__global__ __launch_bounds__(256) void spmm_softmax64(
    const int* __restrict__ rowptr, const v2i* __restrict__ epair,
    const float* __restrict__ Zin, float* __restrict__ out, int nrows) {
  const int lane = threadIdx.x & 31;
  const int wave = threadIdx.x >> 5;
  const int r = blockIdx.x * 8 + wave;
  if (r >= nrows) return;

  v2f acc = (v2f){0.f, 0.f};
  const int e0 = rowptr[r], e1 = rowptr[r + 1];
  for (int e = e0; e < e1; ++e) {
    const v2i   ed = epair[e];
    const float v  = __int_as_float(ed.y);
    const v2f*  z  = (const v2f*)(Zin + (size_t)ed.x * 64);
    acc += v * z[lane];
  }

  float m = fmaxf(acc.x, acc.y);
#pragma unroll
  for (int off = 16; off >= 1; off >>= 1) m = fmaxf(m, __shfl_xor(m, off, 32));
  float x0 = __expf(acc.x - m), x1 = __expf(acc.y - m);
  float s = x0 + x1;
#pragma unroll
  for (int off = 16; off >= 1; off >>= 1) s += __shfl_xor(s, off, 32);
  float inv = 1.f / s;

  v2f res; res.x = x0 * inv; res.y = x1 * inv;
  ((v2f*)(out + (size_t)r * 64))[lane] = res;
}

// ---------------------------------------------------------------------------
extern "C" void kernel_launch(void* const* d_in, const int* in_sizes, int n_in,
                              void* d_out, int out_size, void* d_ws, size_t ws_size,
                              hipStream_t stream) {
  const float* X        = (const float*)d_in[0];
  const int*   edge_row = (const int*)  d_in[1];
  const int*   edge_col = (const int*)  d_in[2];
  const float* edge_val = (const float*)d_in[3];
  const float* W0       = (const float*)d_in[4];
  const float* Wh1      = (const float*)d_in[5];
  const float* W1       = (const float*)d_in[6];
  float* out = (float*)d_out;
  (void)ws_size; (void)n_in; (void)in_sizes; (void)out_size;

  // Workspace layout (~232 MB), 8-byte alignment preserved by ordering:
  float* Hbuf  = (float*)d_ws;                       // N*256 f32
  float* Zbuf  = Hbuf + (size_t)N_NODES * H_DIM;     // N*256 f32
  v2i*   epair = (v2i*)(Zbuf + (size_t)N_NODES * H_DIM);   // E * 8B
  v2f*   Bpack = (v2f*)(epair + N_EDGES);            // up to 64K v2f (512 KB)
  int*   counts = (int*)(Bpack + 65536);             // N
  int*   rowptr = counts + N_NODES;                  // N+1

  const int EB = (N_EDGES + 255) / 256;              // 12500
  const int RB = (N_NODES + 7) / 8;                  // 12500

  // ---- CSR build (shared by all 3 aggregation steps) ----
  hipMemsetAsync(counts, 0, sizeof(int) * N_NODES, stream);
  hist_kernel<<<EB, 256, 0, stream>>>(edge_row, counts, N_EDGES);
  exclusive_scan_kernel<<<1, 1024, 0, stream>>>(counts, rowptr, N_NODES);
  hipMemsetAsync(counts, 0, sizeof(int) * N_NODES, stream);  // reuse as fill
  scatter_kernel<<<EB, 256, 0, stream>>>(edge_row, edge_col, edge_val,
                                         rowptr, counts, epair, N_EDGES);

  // ---- Layer 0: Z0 = relu(A @ (X @ W0)) ----
  {
    int pk = (H_DIM / 16) * (C_DIM / 4) * 32;        // 65536
    packB_kernel<<<(pk + 255) / 256, 256, 0, stream>>>(W0, Bpack, C_DIM, H_DIM);
    dim3 grid(N_NODES / 32, H_DIM / 64);
    gemm_wmma_v2<<<grid, 32, 0, stream>>>(X, Bpack, Hbuf, C_DIM, H_DIM);
    spmm_relu256<<<RB, 256, 0, stream>>>(rowptr, epair, Hbuf, Zbuf, N_NODES);
  }
  // ---- Layer 1: Z1 = relu(A @ (Z0 @ Wh1)) ----
  {
    int pk = (H_DIM / 16) * (H_DIM / 4) * 32;        // 32768
    packB_kernel<<<(pk + 255) / 256, 256, 0, stream>>>(Wh1, Bpack, H_DIM, H_DIM);
    dim3 grid(N_NODES / 32, H_DIM / 64);
    gemm_wmma_v2<<<grid, 32, 0, stream>>>(Zbuf, Bpack, Hbuf, H_DIM, H_DIM);
    spmm_relu256<<<RB, 256, 0, stream>>>(rowptr, epair, Hbuf, Zbuf, N_NODES);
  }
  // ---- Layer 2: out = softmax(A @ (Z1 @ W1)) ----
  {
    int pk = (F_DIM / 16) * (H_DIM / 4) * 32;        // 8192
    packB_kernel<<<(pk + 255) / 256, 256, 0, stream>>>(W1, Bpack, H_DIM, F_DIM);
    dim3 grid(N_NODES / 32, F_DIM / 64);
    gemm_wmma_v2<<<grid, 32, 0, stream>>>(Zbuf, Bpack, Hbuf, H_DIM, F_DIM);
    spmm_softmax64<<<RB, 256, 0, stream>>>(rowptr, epair, Hbuf, out, N_NODES);
  }
}